// VariationalGCNEncoder_609885356342
// MI455X (gfx1250) — compile-verified
//
#include <hip/hip_runtime.h>
#include <hip/hip_bf16.h>

#define N_NODES 50000
#define N_EDGES 800000
#define C_IN    128
#define C_HID   128
#define C_OUT   64
#define BN_EPS  1e-5f

typedef float v2f __attribute__((ext_vector_type(2)));
typedef float v8f __attribute__((ext_vector_type(8)));

// ---------------------------------------------------------------- utilities
__global__ void vgae_zero_kernel(float* __restrict__ p, long n) {
    long i = (long)blockIdx.x * blockDim.x + threadIdx.x;
    long stride = (long)gridDim.x * blockDim.x;
    for (; i < n; i += stride) p[i] = 0.0f;
}

__global__ void vgae_degree_kernel(const int* __restrict__ src, float* __restrict__ deg) {
    int e = blockIdx.x * blockDim.x + threadIdx.x;
    if (e < N_EDGES) atomicAdd(&deg[src[e]], 1.0f);
}

__global__ void vgae_dinv_kernel(const float* __restrict__ deg, float* __restrict__ dinv) {
    int i = blockIdx.x * blockDim.x + threadIdx.x;
    if (i < N_NODES) {
        float d = deg[i];
        dinv[i] = (d > 0.0f) ? rsqrtf(fmaxf(d, 1.0f)) : 0.0f;
    }
}

__global__ void vgae_norm_kernel(const int* __restrict__ src, const int* __restrict__ dst,
                                 const float* __restrict__ dinv, float* __restrict__ nrm) {
    int e = blockIdx.x * blockDim.x + threadIdx.x;
    if (e < N_EDGES) nrm[e] = -dinv[src[e]] * dinv[dst[e]];
}

// One wave per edge; each lane handles 4 of the 128 channels.
// out[dst] += x[src] * norm  via global_atomic_add_f32 (gathers are L2-resident).
__global__ void vgae_scatter_kernel(const float* __restrict__ x,
                                    const int* __restrict__ src, const int* __restrict__ dst,
                                    const float* __restrict__ nrm, float* __restrict__ out) {
    int wave = (blockIdx.x * blockDim.x + threadIdx.x) >> 5;
    int lane = threadIdx.x & 31;
    if (wave >= N_EDGES) return;
    int s = src[wave], d = dst[wave];
    float w = nrm[wave];
    const float4 v = *(const float4*)(x + (long)s * C_HID + lane * 4);
    float* o = out + (long)d * C_HID + lane * 4;
    atomicAdd(o + 0, v.x * w);
    atomicAdd(o + 1, v.y * w);
    atomicAdd(o + 2, v.z * w);
    atomicAdd(o + 3, v.w * w);
}

// ---------------------------------------------------------------- WMMA GEMM
__device__ __forceinline__ v8f vgae_wmma4(v2f a, v2f b, v8f c) {
    return __builtin_amdgcn_wmma_f32_16x16x4_f32(false, a, false, b, (short)0, c,
                                                 false, false);
}

// out = A1 @ W0 + A2 @ W1 + bias.
// Block = 8 waves = COUT/16 column tiles of one row tile; each wave keeps all B
// fragments (both weight matrices, 128 VGPRs) resident and grid-strides over row
// tiles. Two independent accumulator chains break the WMMA RAW dependency.
template <int K>
__global__ __launch_bounds__(256, 1)
void vgae_cheb_gemm_kernel(const float* __restrict__ A1, const float* __restrict__ A2,
                           const float* __restrict__ W0, const float* __restrict__ W1,
                           const float* __restrict__ bias, float* __restrict__ out) {
    constexpr int COUT = 128;
    int wid  = threadIdx.x >> 5;       // 0..7 -> column tile
    int lane = threadIdx.x & 31;
    int half = lane >> 4;
    int l15  = lane & 15;
    int n = wid * 16 + l15;

    v2f bw0[K / 4], bw1[K / 4];
#pragma unroll
    for (int s = 0; s < K / 4; ++s) {
        int kb = 4 * s + 2 * half;
        bw0[s][0] = W0[(kb + 0) * COUT + n];
        bw0[s][1] = W0[(kb + 1) * COUT + n];
        bw1[s][0] = W1[(kb + 0) * COUT + n];
        bw1[s][1] = W1[(kb + 1) * COUT + n];
    }
    float bv = bias[n];

    for (int tm = blockIdx.x; tm < N_NODES / 16; tm += gridDim.x) {
        int m = tm * 16 + l15;
        const float* a1 = A1 + (long)m * K + 2 * half;
        const float* a2 = A2 + (long)m * K + 2 * half;
        int tmn = tm + gridDim.x;
        if (tmn < N_NODES / 16) {          // global_prefetch_b8 of next tile's rows
            __builtin_prefetch(A1 + (long)(tmn * 16 + l15) * K, 0, 1);
            __builtin_prefetch(A2 + (long)(tmn * 16 + l15) * K, 0, 1);
        }
        v8f c0 = {}, c1 = {};
#pragma unroll
        for (int s = 0; s < K / 4; ++s) {
            float2 u = *(const float2*)(a1 + 4 * s);
            float2 w = *(const float2*)(a2 + 4 * s);
            v2f av; av[0] = u.x; av[1] = u.y;
            v2f aw; aw[0] = w.x; aw[1] = w.y;
            c0 = vgae_wmma4(av, bw0[s], c0);
            c1 = vgae_wmma4(aw, bw1[s], c1);
        }
#pragma unroll
        for (int r = 0; r < 8; ++r)
            out[(long)(tm * 16 + r + 8 * half) * COUT + n] = c0[r] + c1[r] + bv;
    }
}

// Fused mu/logstd layer (COUT=64): 8 waves = {mu,logstd} x 4 column tiles;
// one streaming pass over h / tx1h produces both outputs.
template <int K>
__global__ __launch_bounds__(256, 1)
void vgae_cheb_gemm_dual_kernel(const float* __restrict__ A1, const float* __restrict__ A2,
                                const float* __restrict__ Wa0, const float* __restrict__ Wa1,
                                const float* __restrict__ ba,
                                const float* __restrict__ Wb0, const float* __restrict__ Wb1,
                                const float* __restrict__ bb,
                                float* __restrict__ outA, float* __restrict__ outB) {
    constexpr int COUT = 64;
    int wid  = threadIdx.x >> 5;       // 0..7
    int mat  = wid >> 2;               // 0 -> mu, 1 -> logstd
    int tc   = wid & 3;                // column tile
    int lane = threadIdx.x & 31;
    int half = lane >> 4;
    int l15  = lane & 15;
    int n = tc * 16 + l15;

    const float* W0 = mat ? Wb0 : Wa0;
    const float* W1 = mat ? Wb1 : Wa1;
    const float* bp = mat ? bb : ba;
    float* out = mat ? outB : outA;

    v2f bw0[K / 4], bw1[K / 4];
#pragma unroll
    for (int s = 0; s < K / 4; ++s) {
        int kb = 4 * s + 2 * half;
        bw0[s][0] = W0[(kb + 0) * COUT + n];
        bw0[s][1] = W0[(kb + 1) * COUT + n];
        bw1[s][0] = W1[(kb + 0) * COUT + n];
        bw1[s][1] = W1[(kb + 1) * COUT + n];
    }
    float bv = bp[n];

    for (int tm = blockIdx.x; tm < N_NODES / 16; tm += gridDim.x) {
        int m = tm * 16 + l15;
        const float* a1 = A1 + (long)m * K + 2 * half;
        const float* a2 = A2 + (long)m * K + 2 * half;
        int tmn = tm + gridDim.x;
        if (tmn < N_NODES / 16) {
            __builtin_prefetch(A1 + (long)(tmn * 16 + l15) * K, 0, 1);
            __builtin_prefetch(A2 + (long)(tmn * 16 + l15) * K, 0, 1);
        }
        v8f c0 = {}, c1 = {};
#pragma unroll
        for (int s = 0; s < K / 4; ++s) {
            float2 u = *(const float2*)(a1 + 4 * s);
            float2 w = *(const float2*)(a2 + 4 * s);
            v2f av; av[0] = u.x; av[1] = u.y;
            v2f aw; aw[0] = w.x; aw[1] = w.y;
            c0 = vgae_wmma4(av, bw0[s], c0);
            c1 = vgae_wmma4(aw, bw1[s], c1);
        }
#pragma unroll
        for (int r = 0; r < 8; ++r)
            out[(long)(tm * 16 + r + 8 * half) * COUT + n] = c0[r] + c1[r] + bv;
    }
}

// ---------------------------------------------------------------- BatchNorm
// Grid-stride chosen as a multiple of C_HID so each thread owns one channel:
// register accumulation, then one LDS atomic + one global atomic per thread.
__global__ void vgae_bn_stats_kernel(const float* __restrict__ h,
                                     float* __restrict__ sums, float* __restrict__ sqs) {
    __shared__ float ssum[C_HID];
    __shared__ float ssq[C_HID];
    if (threadIdx.x < C_HID) { ssum[threadIdx.x] = 0.0f; ssq[threadIdx.x] = 0.0f; }
    __syncthreads();
    const long total = (long)N_NODES * C_HID;
    long tid = (long)blockIdx.x * blockDim.x + threadIdx.x;
    long stride = (long)gridDim.x * blockDim.x;   // multiple of C_HID by launch config
    int ch = (int)(tid & (C_HID - 1));            // fixed per thread
    float s = 0.0f, q = 0.0f;
    for (long i = tid; i < total; i += stride) {
        float v = h[i];
        s += v;
        q += v * v;
    }
    atomicAdd(&ssum[ch], s);                      // ds_add_f32
    atomicAdd(&ssq[ch], q);
    __syncthreads();
    if (threadIdx.x < C_HID) {
        atomicAdd(&sums[threadIdx.x], ssum[threadIdx.x]);  // global_atomic_add_f32
        atomicAdd(&sqs[threadIdx.x], ssq[threadIdx.x]);
    }
}

__global__ void vgae_bn_relu_kernel(float* __restrict__ h,
                                    const float* __restrict__ sums, const float* __restrict__ sqs,
                                    const float* __restrict__ gamma, const float* __restrict__ beta) {
    const long total = (long)N_NODES * C_HID;
    const float invN = 1.0f / (float)N_NODES;
    long i = (long)blockIdx.x * blockDim.x + threadIdx.x;
    long stride = (long)gridDim.x * blockDim.x;
    for (; i < total; i += stride) {
        int ch = (int)(i & (C_HID - 1));
        float mean = sums[ch] * invN;
        float var = sqs[ch] * invN - mean * mean;
        float v = (h[i] - mean) * rsqrtf(var + BN_EPS) * gamma[ch] + beta[ch];
        h[i] = v > 0.0f ? v : 0.0f;
    }
}

// ---------------------------------------------------------------- launch
extern "C" void kernel_launch(void* const* d_in, const int* in_sizes, int n_in,
                              void* d_out, int out_size, void* d_ws, size_t ws_size,
                              hipStream_t stream) {
    const float* x     = (const float*)d_in[0];
    const int*   ei    = (const int*)d_in[1];
    const float* W1_0  = (const float*)d_in[2];
    const float* W1_1  = (const float*)d_in[3];
    const float* b1    = (const float*)d_in[4];
    const float* gamma = (const float*)d_in[5];
    const float* beta  = (const float*)d_in[6];
    const float* Wmu_0 = (const float*)d_in[7];
    const float* Wmu_1 = (const float*)d_in[8];
    const float* b_mu  = (const float*)d_in[9];
    const float* Wls_0 = (const float*)d_in[10];
    const float* Wls_1 = (const float*)d_in[11];
    const float* b_ls  = (const float*)d_in[12];

    const int* src = ei;            // edge_index[0,:]
    const int* dst = ei + N_EDGES;  // edge_index[1,:]

    // workspace layout (floats); regions needing zero-init come first
    float* ws   = (float*)d_ws;
    float* deg  = ws;                               // N
    float* dinv = deg  + N_NODES;                   // N
    float* nrm  = dinv + N_NODES;                   // E
    float* tx1  = nrm  + N_EDGES;                   // N*C_HID (atomic accum)
    float* tx1h = tx1  + (long)N_NODES * C_HID;     // N*C_HID (atomic accum)
    float* sums = tx1h + (long)N_NODES * C_HID;     // C_HID   (atomic accum)
    float* sqs  = sums + C_HID;                     // C_HID   (atomic accum)
    float* h    = sqs  + C_HID;                     // N*C_HID (fully overwritten)
    long zeroCount = (long)(h - ws);

    float* mu = (float*)d_out;
    float* ls = mu + (long)N_NODES * C_OUT;

    // 1) zero all atomic accumulators
    vgae_zero_kernel<<<2048, 256, 0, stream>>>(ws, zeroCount);
    // 2) degree / dinv / edge norm
    vgae_degree_kernel<<<(N_EDGES + 255) / 256, 256, 0, stream>>>(src, deg);
    vgae_dinv_kernel<<<(N_NODES + 255) / 256, 256, 0, stream>>>(deg, dinv);
    vgae_norm_kernel<<<(N_EDGES + 255) / 256, 256, 0, stream>>>(src, dst, dinv, nrm);
    // 3) tx1 = segment_sum(x[src]*norm, dst)   (one wave per edge)
    vgae_scatter_kernel<<<N_EDGES / 8, 256, 0, stream>>>(x, src, dst, nrm, tx1);
    // 4) h = x @ W1_0 + tx1 @ W1_1 + b1        (register-resident weights)
    vgae_cheb_gemm_kernel<C_IN><<<512, 256, 0, stream>>>(x, tx1, W1_0, W1_1, b1, h);
    // 5) BatchNorm stats + apply + ReLU (in place); grid*block stride % 128 == 0
    vgae_bn_stats_kernel<<<1024, 256, 0, stream>>>(h, sums, sqs);
    vgae_bn_relu_kernel<<<2048, 256, 0, stream>>>(h, sums, sqs, gamma, beta);
    // 6) tx1h = segment_sum(h[src]*norm, dst)
    vgae_scatter_kernel<<<N_EDGES / 8, 256, 0, stream>>>(h, src, dst, nrm, tx1h);
    // 7) fused mu / logstd GEMMs straight into d_out
    vgae_cheb_gemm_dual_kernel<C_HID><<<512, 256, 0, stream>>>(h, tx1h,
                                                               Wmu_0, Wmu_1, b_mu,
                                                               Wls_0, Wls_1, b_ls, mu, ls);
}